// DISCO_NNorm_GAM2_26869315403881
// MI455X (gfx1250) — compile-verified
//
#include <hip/hip_runtime.h>
#include <math.h>

typedef __attribute__((ext_vector_type(2))) float v2f;
typedef __attribute__((ext_vector_type(8))) float v8f;

#define HH    128
#define WWID  256
#define MDIM  66
#define LDIM  64
#define NHEAD 8
#define SDIM  16
#define KSIZ  21
#define JNBR  25
#define HML   32
#define FFNH  256
#define HWPX  (HH*WWID)

__device__ __forceinline__ float gelu_exact(float x) {
    return 0.5f * x * (1.0f + erff(x * 0.70710678118654752f));
}

__device__ __forceinline__ v8f wmma4(v2f a, v2f b, v8f c) {
    // D = A(16x4) * B(4x16) + C, exact fp32
    return __builtin_amdgcn_wmma_f32_16x16x4_f32(false, a, false, b, (short)0, c, false, false);
}

// ---------------- Kernel 1: row-wise FiLM params (gamma/beta) ----------------
__global__ void k_gamma_beta(const float* __restrict__ x,
                             const float* __restrict__ lm_w1, const float* __restrict__ lm_b1,
                             const float* __restrict__ lm_w2, const float* __restrict__ lm_b2,
                             float* __restrict__ gb) {
    int h = threadIdx.x;
    if (h >= HH) return;
    float s0 = x[(size_t)(h * WWID) * MDIM + LDIM + 0];
    float s1 = x[(size_t)(h * WWID) * MDIM + LDIM + 1];
    float g1[32];
    for (int c = 0; c < 32; ++c) {
        float t = s0 * lm_w1[c] + s1 * lm_w1[32 + c] + lm_b1[c];
        g1[c] = gelu_exact(t);
    }
    for (int c2 = 0; c2 < 32; ++c2) {
        float t = lm_b2[c2];
        for (int c = 0; c < 32; ++c) t += g1[c] * lm_w2[c * 32 + c2];
        gb[h * 32 + c2] = t;  // [0:16] gamma, [16:32] beta
    }
}

// ---- Kernel 2: DisCo + FiLM + per-head MLP, all GEMMs on WMMA, 2 pixels/wave ----
__global__ __launch_bounds__(128)
void k_disco_head(const float* __restrict__ x, const int* __restrict__ nbr,
                  const float* __restrict__ psi, const float* __restrict__ disco_w,
                  const float* __restrict__ disco_b,
                  const float* __restrict__ h_w1, const float* __restrict__ h_b1,
                  const float* __restrict__ h_w2, const float* __restrict__ h_b2,
                  const float* __restrict__ gb, float* __restrict__ xl2) {
    __shared__ float wp_lds[8][28][16];   // per-pixel Wp, K padded 25->28 (rows 25..27 = 0)
    __shared__ int   nb_lds[8][28];
    __shared__ float y_lds[8][64][17];    // modulated disco_out (64 ch x 16 s), padded stride

    const int lane  = threadIdx.x & 31;
    const int wid   = threadIdx.x >> 5;
    const int o     = lane & 15;
    const int khalf = 2 * (lane >> 4);
    const int moff  = 8 * (lane >> 4);
    const int p0    = blockIdx.x * 8 + 2 * wid;
    v8f zero = (v8f)0.0f;

    // ---- Phase 1: neighbor indices + Wp = psi_p(28x24 pad) @ disco_w^T(24x16) via WMMA ----
    for (int pix = 0; pix < 2; ++pix) {
        const int slot = 2 * wid + pix;
        const int p = p0 + pix;
        if (lane < 28) nb_lds[slot][lane] = (lane < JNBR) ? nbr[(size_t)p * JNBR + lane] : 0;
        v8f wpa[2];
        wpa[0] = zero; wpa[1] = zero;
        for (int kk = 0; kk < 6; ++kk) {
            int k0  = 4 * kk + khalf;
            int kc0 = (k0     < KSIZ) ? k0     : KSIZ - 1;  // clamp: padded K never reads OOB
            int kc1 = (k0 + 1 < KSIZ) ? k0 + 1 : KSIZ - 1;
            v2f b;                                           // zero B rows kill padded-K terms
            b.x = (k0     < KSIZ) ? disco_w[o * KSIZ + k0]     : 0.f;
            b.y = (k0 + 1 < KSIZ) ? disco_w[o * KSIZ + k0 + 1] : 0.f;
            for (int mt = 0; mt < 2; ++mt) {
                int j  = 16 * mt + (lane & 15);
                int jc = (j < JNBR) ? j : JNBR - 1;          // clamp: padded rows never read OOB
                const float* pr = psi + ((size_t)p * JNBR + jc) * KSIZ;
                v2f a;
                a.x = pr[kc0];
                a.y = pr[kc1];
                wpa[mt] = wmma4(a, b, wpa[mt]);
            }
        }
        for (int mt = 0; mt < 2; ++mt)
            for (int v = 0; v < 8; ++v) {
                int j = 16 * mt + v + moff;
                if (j < 28) wp_lds[slot][j][o] = (j < JNBR) ? wpa[mt][v] : 0.f;
            }
    }
    __syncthreads();

    // ---- Phase 2: y(64x16) = Xg(64x28) @ Wp(28x16), gather straight into A layout ----
    for (int pix = 0; pix < 2; ++pix) {
        const int slot = 2 * wid + pix;
        const int p = p0 + pix;
        v8f acc[4];
        for (int mt = 0; mt < 4; ++mt) acc[mt] = zero;
        for (int kk = 0; kk < 7; ++kk) {
            int k0 = 4 * kk + khalf;
            v2f b;
            b.x = wp_lds[slot][k0][o];
            b.y = wp_lds[slot][k0 + 1][o];
            int q0 = nb_lds[slot][k0];
            int q1 = nb_lds[slot][k0 + 1];
            for (int mt = 0; mt < 4; ++mt) {
                int m = 16 * mt + (lane & 15);
                v2f a;
                a.x = x[(size_t)q0 * MDIM + m];
                a.y = x[(size_t)q1 * MDIM + m];
                acc[mt] = wmma4(a, b, acc[mt]);
            }
        }
        // FiLM modulation on C registers, spill to LDS as y_lds[channel][s]
        int hrow = p >> 8;  // p / W (W=256)
        float g  = gb[hrow * 32 + o];
        float bt = gb[hrow * 32 + 16 + o];
        float db = disco_b[o];
        for (int mt = 0; mt < 4; ++mt)
            for (int v = 0; v < 8; ++v) {
                int m = 16 * mt + v + moff;
                y_lds[slot][m][o] = (acc[mt][v] + db) * g + bt;
            }
    }
    __syncthreads();

    // ---- Phase 3: per-head MLP, layer 1 on WMMA (M = 2 pixels x 8 d = 16 rows) ----
    const int Mrow  = lane & 15;     // A-layout row for this lane
    const int apix  = Mrow >> 3;     // pixel within pair
    const int ad    = Mrow & 7;      // d within head
    const int aslot = 2 * wid + apix;
    for (int n = 0; n < NHEAD; ++n) {
        v8f h1t[2];
        h1t[0] = zero; h1t[1] = zero;
        const int r_in = n * 8 + ad;
        for (int kk = 0; kk < 4; ++kk) {
            int k0 = 4 * kk + khalf;
            v2f a;
            a.x = y_lds[aslot][r_in][k0];
            a.y = y_lds[aslot][r_in][k0 + 1];
            for (int t = 0; t < 2; ++t) {
                int c = 16 * t + (lane & 15);
                v2f b;
                b.x = h_w1[((size_t)n * SDIM + k0) * HML + c];
                b.y = h_w1[((size_t)n * SDIM + k0 + 1) * HML + c];
                h1t[t] = wmma4(a, b, h1t[t]);
            }
        }
        // gelu + layer-2 partials (columns spread across lanes), then 16-lane reduce
        float part[8];
        for (int v = 0; v < 8; ++v) {
            float s = 0.f;
            for (int t = 0; t < 2; ++t) {
                int c = 16 * t + (lane & 15);
                s += gelu_exact(h1t[t][v] + h_b1[n * HML + c]) * h_w2[n * HML + c];
            }
            part[v] = s;
        }
        for (int v = 0; v < 8; ++v) {
            float s = part[v];
            s += __shfl_xor(s, 1);
            s += __shfl_xor(s, 2);
            s += __shfl_xor(s, 4);
            s += __shfl_xor(s, 8);   // stays within each 16-lane half
            part[v] = s;
        }
        // lanes 0 / 16 hold full sums for pixel 0 / 1, rows d = v of head n
        if ((lane & 15) == 0) {
            int p = p0 + (lane >> 4);
            for (int v = 0; v < 8; ++v) {
                int r = n * 8 + v;
                xl2[(size_t)p * LDIM + r] = part[v] + h_b2[n] + x[(size_t)p * MDIM + r];
            }
        }
    }
}

// ---------- Kernel 3: FFN, two chained WMMA GEMMs, one wave per 16 rows ----------
__global__ __launch_bounds__(128)
void k_ffn(const float* __restrict__ x, const float* __restrict__ xl2,
           const float* __restrict__ f_w1, const float* __restrict__ f_b1,
           const float* __restrict__ f_w2, const float* __restrict__ f_b2,
           float* __restrict__ out) {
    __shared__ float hbuf[4][16][260];   // hidden 16x256 per wave, padded stride 260

    const int lane  = threadIdx.x & 31;
    const int wid   = threadIdx.x >> 5;
    const int tile  = blockIdx.x * 4 + wid;
    const int base  = tile * 16;
    const int nlo   = lane & 15;
    const int khalf = 2 * (lane >> 4);
    const int m     = nlo;
    const int row   = base + m;
    const int moff  = 8 * (lane >> 4);
    v8f zero = (v8f)0.0f;

    // Stage A: H = gelu(X(16x66) @ f_w1(66x256) + f_b1), 64-col chunks
    for (int nc = 0; nc < 4; ++nc) {
        v8f a1[4];
        for (int t = 0; t < 4; ++t) a1[t] = zero;
        for (int kk = 0; kk < 16; ++kk) {
            int k0 = 4 * kk + khalf;
            v2f a;
            a.x = xl2[(size_t)row * LDIM + k0];
            a.y = xl2[(size_t)row * LDIM + k0 + 1];
            for (int t = 0; t < 4; ++t) {
                int n = nc * 64 + 16 * t + nlo;
                v2f b;
                b.x = f_w1[(size_t)k0 * FFNH + n];
                b.y = f_w1[(size_t)(k0 + 1) * FFNH + n];
                a1[t] = wmma4(a, b, a1[t]);
            }
        }
        {   // extra K-step: rows 64,65 (sin/cos) of f_w1; rows 66,67 zero-padded
            bool lo = (lane < 16);
            v2f a;
            a.x = lo ? x[(size_t)row * MDIM + 64] : 0.f;
            a.y = lo ? x[(size_t)row * MDIM + 65] : 0.f;
            for (int t = 0; t < 4; ++t) {
                int n = nc * 64 + 16 * t + nlo;
                v2f b;
                b.x = lo ? f_w1[(size_t)64 * FFNH + n] : 0.f;
                b.y = lo ? f_w1[(size_t)65 * FFNH + n] : 0.f;
                a1[t] = wmma4(a, b, a1[t]);
            }
        }
        for (int t = 0; t < 4; ++t) {
            int n = nc * 64 + 16 * t + nlo;
            float bias = f_b1[n];
            for (int v = 0; v < 8; ++v)
                hbuf[wid][v + moff][n] = gelu_exact(a1[t][v] + bias);
        }
    }
    __syncthreads();

    // Stage B: O = H(16x256) @ f_w2(256x64)
    v8f a2[4];
    for (int t = 0; t < 4; ++t) a2[t] = zero;
    for (int kk = 0; kk < 64; ++kk) {
        int k0 = 4 * kk + khalf;
        v2f a;
        a.x = hbuf[wid][m][k0];
        a.y = hbuf[wid][m][k0 + 1];
        for (int t = 0; t < 4; ++t) {
            int n = 16 * t + nlo;
            v2f b;
            b.x = f_w2[(size_t)k0 * LDIM + n];
            b.y = f_w2[(size_t)(k0 + 1) * LDIM + n];
            a2[t] = wmma4(a, b, a2[t]);
        }
    }
    // Epilogue: bias + residual, interleave into 66-channel output, copy sin/cos
    for (int t = 0; t < 4; ++t) {
        int n = 16 * t + nlo;
        float bias = f_b2[n];
        for (int v = 0; v < 8; ++v) {
            int grow = base + v + moff;
            out[(size_t)grow * MDIM + n] = a2[t][v] + bias + xl2[(size_t)grow * LDIM + n];
        }
    }
    if (lane < 16) {
        int grow = base + lane;
        out[(size_t)grow * MDIM + 64] = x[(size_t)grow * MDIM + 64];
        out[(size_t)grow * MDIM + 65] = x[(size_t)grow * MDIM + 65];
    }
}

extern "C" void kernel_launch(void* const* d_in, const int* in_sizes, int n_in,
                              void* d_out, int out_size, void* d_ws, size_t ws_size,
                              hipStream_t stream) {
    const float* x       = (const float*)d_in[0];
    const int*   nbr     = (const int*)  d_in[1];
    const float* psi     = (const float*)d_in[2];
    const float* disco_w = (const float*)d_in[3];
    const float* disco_b = (const float*)d_in[4];
    const float* lm_w1   = (const float*)d_in[5];
    const float* lm_b1   = (const float*)d_in[6];
    const float* lm_w2   = (const float*)d_in[7];
    const float* lm_b2   = (const float*)d_in[8];
    const float* h_w1    = (const float*)d_in[9];
    const float* h_b1    = (const float*)d_in[10];
    const float* h_w2    = (const float*)d_in[11];
    const float* h_b2    = (const float*)d_in[12];
    const float* f_w1    = (const float*)d_in[13];
    const float* f_b1    = (const float*)d_in[14];
    const float* f_w2    = (const float*)d_in[15];
    const float* f_b2    = (const float*)d_in[16];
    float* out = (float*)d_out;

    float* gb  = (float*)d_ws;          // 128*32 floats
    float* xl2 = gb + 128 * 32;         // 32768*64 floats (x_learn2)

    k_gamma_beta<<<1, 128, 0, stream>>>(x, lm_w1, lm_b1, lm_w2, lm_b2, gb);
    k_disco_head<<<HWPX / 8, 128, 0, stream>>>(x, nbr, psi, disco_w, disco_b,
                                               h_w1, h_b1, h_w2, h_b2, gb, xl2);
    k_ffn<<<(HWPX / 16) / 4, 128, 0, stream>>>(x, xl2, f_w1, f_b1, f_w2, f_b2, out);
}